// MaskAttentionBottleneck_74500502716949
// MI455X (gfx1250) — compile-verified
//
#include <hip/hip_runtime.h>
#include <math.h>

typedef __attribute__((ext_vector_type(2))) float v2f;
typedef __attribute__((ext_vector_type(8))) float v8f;

#define BATCH 2
#define CCH   64
#define NVOX  (64*64*64)   /* 262144 voxels per batch */
#define K1CH  8

/* workspace layout (float offsets) */
#define OFF_SPART   0          /* [B][64][K1CH]  = 1024 */
#define OFF_MH      1024       /* [B][64]        = 128  */
#define OFF_PMW     1152       /* [B][64][64]    = 8192 */
#define OFF_PMD     9344       /* [B][64][64]    = 8192 */
#define OFF_QK      17536      /* [B][64]  Wk^T query   */
#define OFF_QV      17664      /* [B][4] : qb, npix, M, npix/Z */
#define OFF_QPEH    17672      /* [B][64] */
#define OFF_QPEW    17800      /* [B][64] */
#define OFF_QPED    17928      /* [B][64] */
#define OFF_LOGITS  18176      /* [B][NVOX] = 524288 (16B aligned) */
#define OFF_PMAX    542464     /* [B][256] block maxima   */
#define OFF_PSUM    542976     /* [B][256] block sum-exp  */

/* ---- k1: s[b,c] = sum_v mask[b,v]*x[b,c,v], split into K1CH chunks ---- */
__global__ void k1_channel_sums(const float* __restrict__ x,
                                const float* __restrict__ mask,
                                float* __restrict__ ws) {
  __shared__ float red[256];
  int blk = blockIdx.x;
  int chunk = blk & (K1CH - 1);
  int c = (blk >> 3) & 63;
  int b = blk >> 9;
  const int vper = NVOX / K1CH;
  size_t xbase = ((size_t)(b * CCH + c)) * NVOX + (size_t)chunk * vper;
  size_t mbase = (size_t)b * NVOX + (size_t)chunk * vper;
  const float4* xv = (const float4*)(x + xbase);
  const float4* mv = (const float4*)(mask + mbase);
  float acc = 0.f;
  for (int i = threadIdx.x; i < vper / 4; i += 256) {
    float4 xx = xv[i];
    float4 mm = mv[i];
    acc += mm.x * xx.x + mm.y * xx.y + mm.z * xx.z + mm.w * xx.w;
  }
  red[threadIdx.x] = acc;
  __syncthreads();
  for (int s = 128; s > 0; s >>= 1) {
    if ((int)threadIdx.x < s) red[threadIdx.x] += red[threadIdx.x + s];
    __syncthreads();
  }
  if (threadIdx.x == 0)
    ws[OFF_SPART + (b * 64 + c) * K1CH + chunk] = red[0];
}

/* ---- k1b: mask marginals m_h[b,h] full; pmw[b,h,w], pmd[b,h,d] partial over h ---- */
__global__ void k1b_marginals(const float* __restrict__ mask, float* __restrict__ ws) {
  __shared__ float part[256];
  __shared__ float tot[256];
  int b = blockIdx.x >> 6;
  int h = blockIdx.x & 63;
  size_t base = (size_t)b * NVOX + (size_t)h * 4096;
  int lane = threadIdx.x & 63;
  int g = threadIdx.x >> 6;       /* 0..3 */
  float s = 0.f;
  for (int d = g * 16; d < g * 16 + 16; ++d) s += mask[base + lane * 64 + d];
  part[lane * 4 + g] = s;
  tot[threadIdx.x] = s;
  __syncthreads();
  if (threadIdx.x < 64) {
    float v = part[threadIdx.x*4] + part[threadIdx.x*4+1] + part[threadIdx.x*4+2] + part[threadIdx.x*4+3];
    ws[OFF_PMW + (b * 64 + h) * 64 + threadIdx.x] = v;
  }
  for (int st = 128; st > 0; st >>= 1) {
    if ((int)threadIdx.x < st) tot[threadIdx.x] += tot[threadIdx.x + st];
    __syncthreads();
  }
  if (threadIdx.x == 0) ws[OFF_MH + b * 64 + h] = tot[0];
  __syncthreads();
  s = 0.f;
  for (int w = g * 16; w < g * 16 + 16; ++w) s += mask[base + w * 64 + lane];
  part[lane * 4 + g] = s;
  __syncthreads();
  if (threadIdx.x < 64) {
    float v = part[threadIdx.x*4] + part[threadIdx.x*4+1] + part[threadIdx.x*4+2] + part[threadIdx.x*4+3];
    ws[OFF_PMD + (b * 64 + h) * 64 + threadIdx.x] = v;
  }
}

/* ---- k2: per-batch combine: query, qk = Wk^T q, qpe tables, qb, npix ---- */
__global__ void k2_combine(const float* __restrict__ Wq, const float* __restrict__ bq,
                           const float* __restrict__ Wk, const float* __restrict__ bk,
                           const float* __restrict__ h_tab, const float* __restrict__ w_tab,
                           const float* __restrict__ d_tab, float* __restrict__ ws) {
  __shared__ float s_[64], mw[64], md[64], mhs[64], query[64];
  int b = blockIdx.x;
  int t = threadIdx.x;  /* 0..63 */
  float a = 0.f;
  for (int k = 0; k < K1CH; ++k) a += ws[OFF_SPART + (b * 64 + t) * K1CH + k];
  s_[t] = a;
  mhs[t] = ws[OFF_MH + b * 64 + t];
  float aw = 0.f, ad = 0.f;
  for (int h = 0; h < 64; ++h) {
    aw += ws[OFF_PMW + (b * 64 + h) * 64 + t];
    ad += ws[OFF_PMD + (b * 64 + h) * 64 + t];
  }
  mw[t] = aw; md[t] = ad;
  __syncthreads();
  float msum = 0.f;
  for (int h = 0; h < 64; ++h) msum += mhs[h];
  float q = bq[t] * msum;
  for (int c = 0; c < 64; ++c) q += Wq[t * 64 + c] * s_[c];
  for (int h = 0; h < 64; ++h)
    q += mhs[h] * h_tab[h * 64 + t] + mw[h] * w_tab[h * 64 + t] + md[h] * d_tab[h * 64 + t];
  q /= msum;
  query[t] = q;
  __syncthreads();
  float qkv = 0.f, ph = 0.f, pw = 0.f, pd = 0.f;
  for (int c = 0; c < 64; ++c) {
    qkv += query[c] * Wk[c * 64 + t];
    ph  += query[c] * h_tab[t * 64 + c];
    pw  += query[c] * w_tab[t * 64 + c];
    pd  += query[c] * d_tab[t * 64 + c];
  }
  ws[OFF_QK   + b * 64 + t] = qkv;
  ws[OFF_QPEH + b * 64 + t] = ph;
  ws[OFF_QPEW + b * 64 + t] = pw;
  ws[OFF_QPED + b * 64 + t] = pd;
  if (t == 0) {
    float qb = 0.f;
    for (int c = 0; c < 64; ++c) qb += query[c] * bk[c];
    ws[OFF_QV + b * 4 + 0] = qb;
    ws[OFF_QV + b * 4 + 1] = msum;
  }
}

/* ---- k3: logits (4 voxels/thread, float4 loads) + fused per-block
        softmax partials (blockmax, sum exp(l - blockmax)) ---- */
__global__ void __launch_bounds__(256) k3_logits(const float* __restrict__ x,
                                                 float* __restrict__ ws) {
  __shared__ float qk_l[64], ph[64], pw[64], pd[64];
  __shared__ float red[256];
  int blk = blockIdx.x;
  int b = blk >> 8;                   /* 256 blocks per batch */
  int bb = blk & 255;
  int vb = bb << 10;                  /* 1024 voxels per block */
  if (threadIdx.x < 64) {
    qk_l[threadIdx.x] = ws[OFF_QK   + b * 64 + threadIdx.x];
    ph[threadIdx.x]   = ws[OFF_QPEH + b * 64 + threadIdx.x];
    pw[threadIdx.x]   = ws[OFF_QPEW + b * 64 + threadIdx.x];
    pd[threadIdx.x]   = ws[OFF_QPED + b * 64 + threadIdx.x];
  }
  __syncthreads();
  int v = vb + threadIdx.x * 4;
  int h = v >> 12, w = (v >> 6) & 63, d0 = v & 63;
  float base = ws[OFF_QV + b * 4 + 0] + ph[h] + pw[w];
  float a0 = base + pd[d0], a1 = base + pd[d0 + 1];
  float a2 = base + pd[d0 + 2], a3 = base + pd[d0 + 3];
  const float4* xp = (const float4*)(x + (size_t)b * CCH * NVOX + v);
#pragma unroll 8
  for (int c = 0; c < 64; ++c) {
    float4 xv = xp[(size_t)c * (NVOX / 4)];
    float qc = qk_l[c];
    a0 += qc * xv.x; a1 += qc * xv.y; a2 += qc * xv.z; a3 += qc * xv.w;
  }
  a0 *= 0.125f; a1 *= 0.125f; a2 *= 0.125f; a3 *= 0.125f;
  *(float4*)(ws + OFF_LOGITS + (size_t)b * NVOX + v) = make_float4(a0, a1, a2, a3);
  /* per-block softmax partials */
  float m = fmaxf(fmaxf(a0, a1), fmaxf(a2, a3));
  red[threadIdx.x] = m;
  __syncthreads();
  for (int s = 128; s > 0; s >>= 1) {
    if ((int)threadIdx.x < s) red[threadIdx.x] = fmaxf(red[threadIdx.x], red[threadIdx.x + s]);
    __syncthreads();
  }
  float bm = red[0];
  __syncthreads();
  float se = __expf(a0 - bm) + __expf(a1 - bm) + __expf(a2 - bm) + __expf(a3 - bm);
  red[threadIdx.x] = se;
  __syncthreads();
  for (int s = 128; s > 0; s >>= 1) {
    if ((int)threadIdx.x < s) red[threadIdx.x] += red[threadIdx.x + s];
    __syncthreads();
  }
  if (threadIdx.x == 0) {
    ws[OFF_PMAX + b * 256 + bb] = bm;
    ws[OFF_PSUM + b * 256 + bb] = red[0];
  }
}

/* ---- k4: combine 256 per-block partials per batch -> M, npix/Z ---- */
__global__ void k4_combine(float* __restrict__ ws) {
  __shared__ float red[256];
  __shared__ float pm[256];
  int b = blockIdx.x, t = threadIdx.x;
  float m = ws[OFF_PMAX + b * 256 + t];
  pm[t] = m;
  red[t] = m;
  __syncthreads();
  for (int s = 128; s > 0; s >>= 1) {
    if (t < s) red[t] = fmaxf(red[t], red[t + s]);
    __syncthreads();
  }
  float M = red[0];
  __syncthreads();
  red[t] = ws[OFF_PSUM + b * 256 + t] * __expf(pm[t] - M);
  __syncthreads();
  for (int s = 128; s > 0; s >>= 1) {
    if (t < s) red[t] += red[t + s];
    __syncthreads();
  }
  if (t == 0) {
    ws[OFF_QV + b * 4 + 2] = M;
    ws[OFF_QV + b * 4 + 3] = ws[OFF_QV + b * 4 + 1] / red[0];
  }
}

/* ---- k5: out = scale(v) * (Wv x + bv) via V_WMMA_F32_16X16X4_F32.
   Per wave: 16 voxels x 64 out-channels = 4 C-tiles x 16 K-steps.
   A (16x4 f32): v0 = K=0|K=2 across lane halves, v1 = K=1|K=3 (ISA layout).
   B (4x16 f32): mirrored: v0 = K=0|K=2, v1 = K=1|K=3. ---- */
__global__ void __launch_bounds__(256) k5_vals_wmma(const float* __restrict__ x,
                                                    const float* __restrict__ Wv,
                                                    const float* __restrict__ bv,
                                                    const float* __restrict__ ws,
                                                    float* __restrict__ out) {
  __shared__ float wv_l[64 * 64];
  __shared__ float bv_l[64];
  int blk = blockIdx.x;
  int b = blk >> 11;                  /* 2048 blocks per batch */
  int vbase = (blk & 2047) << 7;      /* 128 voxels per block  */
  for (int i = threadIdx.x; i < 4096; i += 256) wv_l[i] = Wv[i];
  if (threadIdx.x < 64) bv_l[threadIdx.x] = bv[threadIdx.x];
  __syncthreads();

  int lane  = threadIdx.x & 31;
  int wave  = threadIdx.x >> 5;
  int half  = lane >> 4;
  int col16 = lane & 15;
  int vcol  = vbase + wave * 16 + col16;

  /* B operand registers: 64 channels x this lane's voxel column */
  float b0r[16], b1r[16];
  size_t xb = (size_t)b * CCH * NVOX + vcol;
#pragma unroll
  for (int kk = 0; kk < 16; ++kk) {
    int c0 = 4 * kk + (half ? 2 : 0);
    b0r[kk] = x[xb + (size_t)c0 * NVOX];
    b1r[kk] = x[xb + (size_t)(c0 + 1) * NVOX];
  }

  float M     = ws[OFF_QV + b * 4 + 2];
  float noz   = ws[OFF_QV + b * 4 + 3];
  float scale = noz * __expf(ws[OFF_LOGITS + (size_t)b * NVOX + vcol] - M);

#pragma unroll
  for (int mt = 0; mt < 4; ++mt) {
    v8f acc;
#pragma unroll
    for (int j = 0; j < 8; ++j) acc[j] = bv_l[mt * 16 + j + half * 8];
#pragma unroll
    for (int kk = 0; kk < 16; ++kk) {
      int arow = mt * 16 + col16;
      int acol = 4 * kk + (half ? 2 : 0);
      v2f a;  a[0]  = wv_l[arow * 64 + acol];
              a[1]  = wv_l[arow * 64 + acol + 1];
      v2f bb; bb[0] = b0r[kk];
              bb[1] = b1r[kk];
      acc = __builtin_amdgcn_wmma_f32_16x16x4_f32(false, a, false, bb,
                                                  (short)0, acc, false, false);
    }
    size_t ob = ((size_t)b * CCH + mt * 16 + half * 8) * NVOX + vcol;
#pragma unroll
    for (int j = 0; j < 8; ++j) out[ob + (size_t)j * NVOX] = acc[j] * scale;
  }
}

extern "C" void kernel_launch(void* const* d_in, const int* in_sizes, int n_in,
                              void* d_out, int out_size, void* d_ws, size_t ws_size,
                              hipStream_t stream) {
  const float* x     = (const float*)d_in[0];
  const float* mask  = (const float*)d_in[1];
  const float* Wk    = (const float*)d_in[2];
  const float* bk    = (const float*)d_in[3];
  const float* Wv    = (const float*)d_in[4];
  const float* bv    = (const float*)d_in[5];
  const float* Wq    = (const float*)d_in[6];
  const float* bq    = (const float*)d_in[7];
  const float* h_tab = (const float*)d_in[8];
  const float* w_tab = (const float*)d_in[9];
  const float* d_tab = (const float*)d_in[10];
  float* out = (float*)d_out;
  float* ws  = (float*)d_ws;

  k1_channel_sums<<<BATCH * 64 * K1CH,  256, 0, stream>>>(x, mask, ws);
  k1b_marginals  <<<BATCH * 64,         256, 0, stream>>>(mask, ws);
  k2_combine     <<<BATCH,               64, 0, stream>>>(Wq, bq, Wk, bk, h_tab, w_tab, d_tab, ws);
  k3_logits      <<<BATCH * (NVOX/1024), 256, 0, stream>>>(x, ws);
  k4_combine     <<<BATCH,              256, 0, stream>>>(ws);
  k5_vals_wmma   <<<BATCH * (NVOX/128), 256, 0, stream>>>(x, Wv, bv, ws, out);

  /* output tuple tail: the mask passthrough */
  hipMemcpyAsync(out + (size_t)BATCH * CCH * NVOX, mask,
                 (size_t)BATCH * NVOX * sizeof(float),
                 hipMemcpyDeviceToDevice, stream);
}